// GridMask_25658134626696
// MI455X (gfx1250) — compile-verified
//
#include <hip/hip_runtime.h>

// GridMask: out[b,y,x,c] = x[b,y,x,c] * keep(b,y,x)
// keep = !( ((OFF+y - st_h) mod d) < l  ||  ((OFF+x - st_w) mod d) < l )
// d = 96 + d_raw[b], l = ceil(d/2), st_* = st_*_raw[b] mod d, OFF = 106.
//
// Pure streaming kernel: 201 MB total traffic, ~8.6 us at 23.3 TB/s.
// Optimizations:
//  * 128-bit (float4) loads/stores, nontemporal (stream > 192 MB L2).
//  * Mask computed BEFORE the load; ~75% of pixels are zero, so waves whose
//    entire 128-float span is masked skip the global load and store zeros.
//  * Runtime-d modulo via float-reciprocal fast-mod (values bounded by 617,
//    f32 is exact enough; one conditional +/-d fixup).
//  * C=3 means a float4 touches only two pixels: one column fast-mod + an
//    incremental "+1 with wrap" yields both column masks.

typedef float v4f __attribute__((ext_vector_type(4)));

namespace {
constexpr int Bc  = 32;
constexpr int Hc  = 512;
constexpr int Wc  = 512;
constexpr int Cc  = 3;
constexpr int ROW = Wc * Cc;        // 1536 floats per image row
constexpr int IMG = Hc * ROW;       // 786432 floats per image
constexpr int OFF = 106;            // (ceil(sqrt(2)*512)=725 - 512)/2
constexpr int D1c = 96;
constexpr int THREADS = 256;
}

// fast a mod d for a in [-212, 617], d in [96, 320)
__device__ __forceinline__ int fastmod(int a, int d, float rcp_d) {
    int q = (int)floorf((float)a * rcp_d);
    int r = a - q * d;
    r += (r < 0)  ? d : 0;
    r -= (r >= d) ? d : 0;
    return r;
}

__global__ __launch_bounds__(THREADS)
void gridmask_kernel(const float* __restrict__ x,
                     const int*   __restrict__ d_raw,
                     const int*   __restrict__ st_h_raw,
                     const int*   __restrict__ st_w_raw,
                     float*       __restrict__ out) {
    // ---- block-uniform per-image parameters (scalar loads / SALU) ----
    const int   b   = blockIdx.y;
    const int   d   = D1c + d_raw[b];
    const int   l   = (d + 1) >> 1;          // ceil(d * 0.5)
    const int   sh  = st_h_raw[b] % d;
    const int   sw  = st_w_raw[b] % d;
    const float rcp = 1.0f / (float)d;

    // ---- this thread's 4-float chunk inside image b ----
    const int off = (blockIdx.x * THREADS + (int)threadIdx.x) * 4;  // [0, IMG)
    const int y   = (int)((unsigned)off / (unsigned)ROW);           // const divisor
    const int rr  = off - y * ROW;                                  // [0, 1536)

    // row stripe test (same for all 4 elements)
    const int  rrow = fastmod(y + (OFF - sh), d, rcp);
    const bool rz   = rrow < l;

    // the float4 covers pixels p0 and p0+1 only (C == 3)
    const int p0 = (int)((unsigned)rr / 3u);
    const int t  = rr - p0 * 3;                                     // {0,1,2}
    const int c0 = fastmod(p0 + (OFF - sw), d, rcp);
    int c1 = c0 + 1; c1 = (c1 >= d) ? 0 : c1;

    const bool k0 = (!rz) && (c0 >= l);      // keep pixel p0
    const bool k1 = (!rz) && (c1 >= l);      // keep pixel p0+1

    const size_t g = (size_t)b * (size_t)IMG + (size_t)off;

    v4f o;
    if (__any((int)(k0 | k1))) {
        // at least one lane in the wave keeps data -> do the 128-bit load
        const v4f v = __builtin_nontemporal_load((const v4f*)(x + g));
        const bool m0 = k0;                   // t+0 < 3 always
        const bool m1 = (t < 2) ? k0 : k1;
        const bool m2 = (t < 1) ? k0 : k1;
        const bool m3 = k1;                   // t+3 >= 3 always
        o[0] = m0 ? v[0] : 0.0f;
        o[1] = m1 ? v[1] : 0.0f;
        o[2] = m2 ? v[2] : 0.0f;
        o[3] = m3 ? v[3] : 0.0f;
    } else {
        // whole wave masked: skip the read entirely (saves ~60-75% of read BW)
        o = (v4f){0.0f, 0.0f, 0.0f, 0.0f};
    }
    __builtin_nontemporal_store(o, (v4f*)(out + g));
}

extern "C" void kernel_launch(void* const* d_in, const int* in_sizes, int n_in,
                              void* d_out, int out_size, void* d_ws, size_t ws_size,
                              hipStream_t stream) {
    (void)in_sizes; (void)n_in; (void)d_ws; (void)ws_size; (void)out_size;
    const float* x        = (const float*)d_in[0];
    const int*   d_raw    = (const int*)  d_in[1];
    const int*   st_h_raw = (const int*)  d_in[2];
    const int*   st_w_raw = (const int*)  d_in[3];
    float*       out      = (float*)d_out;

    dim3 grid(IMG / (THREADS * 4), Bc);   // (768, 32)
    gridmask_kernel<<<grid, THREADS, 0, stream>>>(x, d_raw, st_h_raw, st_w_raw, out);
}